// densityDiffusionModule_55482387530424
// MI455X (gfx1250) — compile-verified
//
#include <hip/hip_runtime.h>

// ---------------- constants (from reference) ----------------
#define H_      0.05f
#define INV_H2  400.0f        // 1/h^2
#define INV_H3  8000.0f       // 1/h^3
#define KC_     2.2281692032865347f   // 7/pi
#define EPS_    2.5e-4f       // h^2 * 0.1
#define REST_   1000.0f
#define SCALE_  0.2426108f    // 2*h*DELTA*C0 = 0.01 * 10*sqrt(2*9.81*0.3)
#define EPS_DW  5.0e-6f       // 1e-4 * h

#define CHUNK 8
#define BLK   256
// 176B slot (44 dwords): dir 64 | q 32 | i 32 | j 32 | pad 16
// stride 44 dwords mod 64 banks -> 2-way conflict (vs 4-way at 160B)
#define SLOT  176

// ---------------- CDNA5 async global->LDS helpers ----------------
__device__ __forceinline__ unsigned lds_off_of(const void* p) {
  // generic LDS pointer: addr[31:0] is the wave-relative LDS byte offset (ISA 10.2)
  return (unsigned)(unsigned long long)p;
}
__device__ __forceinline__ void async_b128(unsigned lds_off, const void* g) {
  // non-temporal: edge streams are read-once, keep them from thrashing L2
  asm volatile("global_load_async_to_lds_b128 %0, %1, off th:TH_LOAD_NT"
               :: "v"(lds_off), "v"(g) : "memory");
}
__device__ __forceinline__ void wait_async0() {
  asm volatile("s_wait_asynccnt 0" ::: "memory");
}

__device__ __forceinline__ void fadd_atomic(float* p, float v) {
  unsafeAtomicAdd(p, v);     // global_atomic_add_f32
}

// ---------------- per-thread edge tile ----------------
struct EdgeRegs {
  float dx[CHUNK], dy[CHUNK], q[CHUNK];
  int   iv[CHUNK], jv[CHUNK];
};

__device__ __forceinline__ void load_edges(
    const float* __restrict__ dirs, const float* __restrict__ rad,
    const int* __restrict__ ii, const int* __restrict__ jj,
    int E, char* smem, EdgeRegs& r)
{
  const int tid      = threadIdx.x;
  const int tileBase = (int)blockIdx.x * (BLK * CHUNK);
  const int base     = tileBase + tid * CHUNK;
  const bool full    = (tileBase + BLK * CHUNK) <= E;   // uniform per block

  if (full) {
    char* slot = smem + tid * SLOT;
    unsigned so = lds_off_of(slot);
    const char* gd = (const char*)dirs + (size_t)base * 8u;   // 64B of dirs
    async_b128(so +   0, gd +  0);
    async_b128(so +  16, gd + 16);
    async_b128(so +  32, gd + 32);
    async_b128(so +  48, gd + 48);
    const char* gq = (const char*)rad + (size_t)base * 4u;    // 32B of q
    async_b128(so +  64, gq +  0);
    async_b128(so +  80, gq + 16);
    const char* gi = (const char*)ii + (size_t)base * 4u;     // 32B of i
    async_b128(so +  96, gi +  0);
    async_b128(so + 112, gi + 16);
    const char* gj = (const char*)jj + (size_t)base * 4u;     // 32B of j
    async_b128(so + 128, gj +  0);
    async_b128(so + 144, gj + 16);
    wait_async0();                      // each thread reads only its own slot
    const float* sd = (const float*)(slot);
    const float* sq = (const float*)(slot + 64);
    const int*   si = (const int*)  (slot + 96);
    const int*   sj = (const int*)  (slot + 128);
#pragma unroll
    for (int c = 0; c < CHUNK; ++c) {
      r.dx[c] = sd[2*c]; r.dy[c] = sd[2*c+1];
      r.q[c]  = sq[c];   r.iv[c] = si[c];   r.jv[c] = sj[c];
    }
  } else {
#pragma unroll
    for (int c = 0; c < CHUNK; ++c) {
      int e = base + c;
      if (e < E) {
        r.dx[c] = dirs[2*e]; r.dy[c] = dirs[2*e+1];
        r.q[c]  = rad[e];    r.iv[c] = ii[e];       r.jv[c] = jj[e];
      } else {
        r.iv[c] = -1; r.jv[c] = 0; r.dx[c] = 0.f; r.dy[c] = 0.f; r.q[c] = 0.f;
      }
    }
  }
}

// ---------------- pass 0: zero workspace + output ----------------
__global__ void zero_kernel(float* ws, float* out, int nws, int nout) {
  int t = blockIdx.x * blockDim.x + threadIdx.x;
  if (t < nws)  ws[t]  = 0.f;
  if (t < nout) out[t] = 0.f;
}

// ---------------- pass A: M[i] += 2 Vj (r_ba ⊗ gradW)  (symmetric) ----------------
__global__ void __launch_bounds__(BLK) pass_M(
    const float* __restrict__ vol, const float* __restrict__ dirs,
    const float* __restrict__ rad, const int* __restrict__ ii,
    const int* __restrict__ jj, float* __restrict__ Msym, int E)
{
  __shared__ __align__(16) char smem[BLK * SLOT];
  EdgeRegs r;
  load_edges(dirs, rad, ii, jj, E, smem, r);

  int cur = -1; float axx = 0.f, axy = 0.f, ayy = 0.f;
#pragma unroll
  for (int c = 0; c < CHUNK; ++c) {
    int ie = r.iv[c];
    if (ie < 0) continue;
    if (ie != cur) {
      if (cur >= 0) {
        fadd_atomic(&Msym[3*cur+0], axx);
        fadd_atomic(&Msym[3*cur+1], axy);
        fadd_atomic(&Msym[3*cur+2], ayy);
      }
      cur = ie; axx = axy = ayy = 0.f;
    }
    float q = r.q[c], dx = r.dx[c], dy = r.dy[c];
    float Vj  = vol[r.jv[c]];
    float omq = 1.f - q;
    float dkdq = -20.f * q * omq * omq * omq * KC_;
    // r_ba[u]*gradW[v] = (-q*dkdq/h^2) * dir_u * dir_v  (PSD, symmetric)
    float s = -2.f * Vj * q * dkdq * INV_H2;
    axx += s * dx * dx; axy += s * dx * dy; ayy += s * dy * dy;
  }
  if (cur >= 0) {
    fadd_atomic(&Msym[3*cur+0], axx);
    fadd_atomic(&Msym[3*cur+1], axy);
    fadd_atomic(&Msym[3*cur+2], ayy);
  }
}

// ---------------- pass B: symmetric 2x2 pinv, in place ----------------
__global__ void pass_pinv(float* __restrict__ Msym, int N) {
  int t = blockIdx.x * blockDim.x + threadIdx.x;
  if (t >= N) return;
  float mxx = Msym[3*t], mxy = Msym[3*t+1], myy = Msym[3*t+2];
  float tr  = mxx + myy;
  float det = mxx * myy - mxy * mxy;
  float disc = sqrtf(fmaxf(tr*tr - 4.f*det, 0.f));
  float l1 = 0.5f * (tr + disc);          // eigen = singular values (PSD)
  float l2 = 0.5f * (tr - disc);
  const float rcond = 2.3841858e-6f;      // 10*max(M,N)*eps_f32 (JAX default)
  float lxx, lxy, lyy;
  if (l2 > rcond * l1 && det != 0.f) {    // full rank: pinv == inverse
    float idet = 1.f / det;
    lxx =  myy * idet; lxy = -mxy * idet; lyy =  mxx * idet;
  } else if (l1 > 0.f) {                  // rank-1: pinv ≈ A / l1^2
    float s = 1.f / (l1 * l1);
    lxx = mxx * s; lxy = mxy * s; lyy = myy * s;
  } else {
    lxx = lxy = lyy = 0.f;
  }
  Msym[3*t] = lxx; Msym[3*t+1] = lxy; Msym[3*t+2] = lyy;
}

// ---------------- pass C: gradRho[i] += 2 (rho_j-rho_i) Vj * grad ----------------
__global__ void __launch_bounds__(BLK) pass_gradRho(
    const float* __restrict__ vol, const float* __restrict__ dens,
    const float* __restrict__ dirs, const float* __restrict__ rad,
    const int* __restrict__ ii, const int* __restrict__ jj,
    const float* __restrict__ Li, float* __restrict__ gradRho, int E)
{
  __shared__ __align__(16) char smem[BLK * SLOT];
  EdgeRegs r;
  load_edges(dirs, rad, ii, jj, E, smem, r);

  int cur = -1; float gx = 0.f, gy = 0.f;
  float lxx = 0.f, lxy = 0.f, lyy = 0.f, rho_i = 0.f;
#pragma unroll
  for (int c = 0; c < CHUNK; ++c) {
    int ie = r.iv[c];
    if (ie < 0) continue;
    if (ie != cur) {
      if (cur >= 0) {
        fadd_atomic(&gradRho[2*cur+0], gx);
        fadd_atomic(&gradRho[2*cur+1], gy);
      }
      cur = ie; gx = gy = 0.f;
      lxx = Li[3*ie]; lxy = Li[3*ie+1]; lyy = Li[3*ie+2];
      rho_i = dens[ie] * REST_;
    }
    float q = r.q[c], dx = r.dx[c], dy = r.dy[c];
    float omq = 1.f - q;
    float dkdq = -20.f * q * omq * omq * omq * KC_;
    float gw = dkdq * INV_H3;
    float g0 = dx * gw, g1 = dy * gw;
    float p0 = lxx * g0 + lxy * g1;
    float p1 = lxy * g0 + lyy * g1;
    float dwmag = fabsf(g0) + fabsf(g1);
    float nmag  = fabsf(p0) + fabsf(p1);
    float change = fabsf(nmag - dwmag) / (dwmag + EPS_DW);
    bool ok = change < 0.1f;
    float gr0 = ok ? p0 : g0;
    float gr1 = ok ? p1 : g1;
    int jn = r.jv[c];
    float w = 2.f * (dens[jn] * REST_ - rho_i) * vol[jn];
    gx += w * gr0; gy += w * gr1;
  }
  if (cur >= 0) {
    fadd_atomic(&gradRho[2*cur+0], gx);
    fadd_atomic(&gradRho[2*cur+1], gy);
  }
}

// ---------------- pass D: final delta-SPH diffusion scatter ----------------
__global__ void __launch_bounds__(BLK) pass_out(
    const float* __restrict__ vol, const float* __restrict__ dens,
    const float* __restrict__ dirs, const float* __restrict__ rad,
    const int* __restrict__ ii, const int* __restrict__ jj,
    const float* __restrict__ gradRho, float* __restrict__ out, int E)
{
  __shared__ __align__(16) char smem[BLK * SLOT];
  EdgeRegs r;
  load_edges(dirs, rad, ii, jj, E, smem, r);

  int cur = -1; float acc = 0.f;
  float gRix = 0.f, gRiy = 0.f, rho_i = 0.f;
#pragma unroll
  for (int c = 0; c < CHUNK; ++c) {
    int ie = r.iv[c];
    if (ie < 0) continue;
    if (ie != cur) {
      if (cur >= 0) fadd_atomic(&out[cur], SCALE_ * acc);
      cur = ie; acc = 0.f;
      gRix = gradRho[2*ie]; gRiy = gradRho[2*ie+1];
      rho_i = dens[ie] * REST_;
    }
    float q = r.q[c], dx = r.dx[c], dy = r.dy[c];
    float omq = 1.f - q;
    float dkdq = -20.f * q * omq * omq * omq * KC_;
    float gw = dkdq * INV_H3;
    float g0 = dx * gw, g1 = dy * gw;
    float rx = -dx * q * H_, ry = -dy * q * H_;
    float rji2 = rx * rx + ry * ry + EPS_;
    int jn = r.jv[c];
    float gRjx = gradRho[2*jn], gRjy = gradRho[2*jn+1];
    float densityTerm = 0.5f * ((gRix + gRjx) * rx + (gRiy + gRjy) * ry);
    float gradTerm = (g0 * rx + g1 * ry) / rji2;
    float rho_ba = dens[jn] * REST_ - rho_i;
    acc += (rho_ba + densityTerm) * gradTerm * vol[jn];
  }
  if (cur >= 0) fadd_atomic(&out[cur], SCALE_ * acc);
}

// ---------------- launcher ----------------
extern "C" void kernel_launch(void* const* d_in, const int* in_sizes, int n_in,
                              void* d_out, int out_size, void* d_ws, size_t ws_size,
                              hipStream_t stream) {
  (void)n_in; (void)out_size; (void)ws_size;
  const float* vol  = (const float*)d_in[1];
  const float* dens = (const float*)d_in[2];
  const float* dirs = (const float*)d_in[3];
  const float* rad  = (const float*)d_in[4];
  const int*   ii   = (const int*)d_in[5];
  const int*   jj   = (const int*)d_in[6];
  const int N = in_sizes[1];
  const int E = in_sizes[4];

  float* out     = (float*)d_out;
  float* Msym    = (float*)d_ws;                 // 3N floats: M then Li (in place)
  float* gradRho = Msym + 3 * (size_t)N;         // 2N floats
  const int nws = 5 * N;

  zero_kernel<<<(nws + 255) / 256, 256, 0, stream>>>(Msym, out, nws, N);

  const int eThreads = (E + CHUNK - 1) / CHUNK;
  const int eBlocks  = (eThreads + BLK - 1) / BLK;

  pass_M      <<<eBlocks, BLK, 0, stream>>>(vol, dirs, rad, ii, jj, Msym, E);
  pass_pinv   <<<(N + 255) / 256, 256, 0, stream>>>(Msym, N);
  pass_gradRho<<<eBlocks, BLK, 0, stream>>>(vol, dens, dirs, rad, ii, jj, Msym, gradRho, E);
  pass_out    <<<eBlocks, BLK, 0, stream>>>(vol, dens, dirs, rad, ii, jj, gradRho, out, E);
}